// ConvCapsule_88046829568618
// MI455X (gfx1250) — compile-verified
//
#include <hip/hip_runtime.h>
#include <hip/hip_bf16.h>
#include <math.h>

// Shapes (from reference): input [16,32,32,16,8] f32, W [3,3,8,128] f32, b [1,1,16,8] f32.
// Output activation [16,32,32,16,8] f32 (axis-0 is the reference's swapped axis).

typedef __attribute__((ext_vector_type(2))) float v2f;
typedef __attribute__((ext_vector_type(8))) float v8f;

#define NPIX      16384      // 16 * 32 * 32
#define KDIM      72         // 3*3*8
#define NDIM      128        // 16 caps * 8 dims
#define LDSW_STR  132        // padded row stride for W in LDS (bank-conflict avoidance)
#define VOTE_STR  130        // padded row stride for votes in LDS
#define CAPS_EPS  1e-7f
#define PH        34         // padded spatial (zero halo for SAME conv)
#define PLANE     (PH * PH * 128)   // floats per ic-plane in padded buffer
#define PIX_PER_WAVE 4

// ---------------------------------------------------------------------------
// Pre-pass: input [b,h,w,ic,id] -> xTp [ic, h+1, w+1, b, id] with a zero halo,
// so (a) the per-pixel GEMM's M dimension (batch, due to the reference axis
// quirk) is contiguous and (b) A-fragment loads need no bounds checks.
// ---------------------------------------------------------------------------
__global__ void caps_pad_transpose_kernel(const float* __restrict__ in,
                                          float* __restrict__ out) {
    int t = blockIdx.x * blockDim.x + threadIdx.x;   // over 16*34*34*16 groups of 8 floats
    if (t >= 16 * PH * PH * 16) return;
    int b  = t & 15;
    int r  = t >> 4;
    int ww = r % PH;  r /= PH;
    int hh = r % PH;
    int ic = r / PH;
    int h = hh - 1;
    int w = ww - 1;
    float4 v0 = float4{0.f, 0.f, 0.f, 0.f};
    float4 v1 = v0;
    if ((unsigned)h < 32u && (unsigned)w < 32u) {
        const float4* src = (const float4*)(in + ((((size_t)b * 32 + h) * 32 + w) * 16 + ic) * 8);
        v0 = src[0];
        v1 = src[1];
    }
    float4* dst = (float4*)(out + (size_t)t * 8);    // coalesced writes
    dst[0] = v0;
    dst[1] = v1;
}

// ---------------------------------------------------------------------------
// Fused implicit-GEMM conv (fp32 WMMA 16x16x4) + 2-iteration dynamic routing.
// One wave handles PIX_PER_WAVE consecutive output pixels, reusing W staged in
// LDS across pixels.
// ---------------------------------------------------------------------------
__global__ __launch_bounds__(64) void caps_conv_route_kernel(
    const float* __restrict__ xTp,    // [16,34,34,16,8]  (ic, h+1, w+1, b, id), zero halo
    const float* __restrict__ Wg,     // [72,128]
    const float* __restrict__ bias,   // [128]
    float* __restrict__ out)          // [16,32,32,128]
{
    __shared__ float ldsW[KDIM * LDSW_STR];
    __shared__ float ldsVotes[2][16 * VOTE_STR];
    __shared__ float ldsLogits[2][16 * 16];
    __shared__ float ldsMx[2][16];
    __shared__ float ldsRs[2][16];

    const int tid  = threadIdx.x;     // 0..63
    const int wib  = tid >> 5;        // wave in block: 0..1
    const int lane = tid & 31;

    // Stage weight matrix [72][128] into padded LDS.
    for (int i = tid; i < KDIM * NDIM; i += 64) {
        int r = i >> 7;
        int c = i & 127;
        ldsW[r * LDSW_STR + c] = Wg[i];
    }
    __syncthreads();

    const int m  = lane & 15;   // GEMM row = true batch index j
    const int hi = lane >> 4;   // half-wave selector (K sub-block)

    // Per-lane bias values (constant across the pixel loop).
    float bias_l[4];
#pragma unroll
    for (int q = 0; q < 4; ++q) bias_l[q] = bias[q * 32 + lane];

    float* votes  = ldsVotes[wib];
    float* logits = ldsLogits[wib];

    const int waveId = blockIdx.x * 2 + wib;

#pragma unroll 1
    for (int it = 0; it < PIX_PER_WAVE; ++it) {
        const int pixel = waveId * PIX_PER_WAVE + it;   // 0..16383
        const int icap  = pixel >> 10;    // output axis-0 == input capsule (ref quirk)
        const int h     = (pixel >> 5) & 31;
        const int w     = pixel & 31;

        v8f acc[8];
#pragma unroll
        for (int t = 0; t < 8; ++t)
            acc[t] = v8f{0.f, 0.f, 0.f, 0.f, 0.f, 0.f, 0.f, 0.f};

        // Halo origin for this pixel: padded (h,w) is the (kh=0,kw=0) tap.
        const float* base = xTp + (size_t)icap * PLANE + (h * PH + w) * 128;

#pragma unroll
        for (int ks = 0; ks < 18; ++ks) {
            // A fragment: 16x4 fp32 layout. lanes 0-15 hold K=4ks+{0,1}; lanes 16-31 K=4ks+{2,3}
            int K0  = ks * 4 + hi * 2;
            int kh  = K0 / 24;
            int rem = K0 - kh * 24;
            int kw  = rem >> 3;
            int id0 = rem & 7;            // even -> pair stays within one (kh,kw) tap
            // Unconditional, coalesced 8B load (halo supplies the zeros).
            v2f a = *(const v2f*)(base + (kh * PH + kw) * 128 + m * 8 + id0);
#pragma unroll
            for (int t = 0; t < 8; ++t) {
                // B fragment: rows K0, K0+1, column tile*16 + n (n = lane&15)
                int col = t * 16 + m;
                v2f bf;
                bf[0] = ldsW[K0 * LDSW_STR + col];
                bf[1] = ldsW[(K0 + 1) * LDSW_STR + col];
                acc[t] = __builtin_amdgcn_wmma_f32_16x16x4_f32(
                    false, a, false, bf, (short)0, acc[t], false, false);
            }
        }

        // Spill votes [M=16][C=128] to per-wave LDS slab.
        // C/D layout: VGPR v -> M = v (lanes 0-15) or v+8 (lanes 16-31); N = lane&15.
#pragma unroll
        for (int t = 0; t < 8; ++t)
#pragma unroll
            for (int v = 0; v < 8; ++v)
                votes[(v + 8 * hi) * VOTE_STR + t * 16 + m] = acc[t][v];

        // ---- Routing iteration 1: route = 1/16 uniform ----
        float pre[4], act[4];
#pragma unroll
        for (int q = 0; q < 4; ++q) {
            int c = q * 32 + lane;
            float s = 0.f;
            for (int ic = 0; ic < 16; ++ic) s += votes[ic * VOTE_STR + c];
            pre[q] = s * (1.f / 16.f) + bias_l[q];
        }
#pragma unroll
        for (int q = 0; q < 4; ++q) {
            float n2 = pre[q] * pre[q];
            n2 += __shfl_xor(n2, 1, 32);
            n2 += __shfl_xor(n2, 2, 32);
            n2 += __shfl_xor(n2, 4, 32);
            float scale = (n2 / (1.f + n2)) / sqrtf(n2 + CAPS_EPS);
            act[q] = pre[q] * scale;
        }

        // Agreement -> logits[ic][oc]
        for (int ic = 0; ic < 16; ++ic) {
#pragma unroll
            for (int q = 0; q < 4; ++q) {
                float d = votes[ic * VOTE_STR + q * 32 + lane] * act[q];
                d += __shfl_xor(d, 1, 32);
                d += __shfl_xor(d, 2, 32);
                d += __shfl_xor(d, 4, 32);
                if ((lane & 7) == 0)
                    logits[ic * 16 + q * 4 + (lane >> 3)] = d;
            }
        }

        // Per-row softmax statistics.
        if (lane < 16) {
            float mx = -3.0e38f;
            for (int oc = 0; oc < 16; ++oc) mx = fmaxf(mx, logits[lane * 16 + oc]);
            float s = 0.f;
            for (int oc = 0; oc < 16; ++oc) s += __expf(logits[lane * 16 + oc] - mx);
            ldsMx[wib][lane] = mx;
            ldsRs[wib][lane] = 1.f / s;
        }

        // ---- Routing iteration 2 ----
#pragma unroll
        for (int q = 0; q < 4; ++q) {
            int c  = q * 32 + lane;
            int oc = c >> 3;
            float s = 0.f;
            for (int ic = 0; ic < 16; ++ic) {
                float r = __expf(logits[ic * 16 + oc] - ldsMx[wib][ic]) * ldsRs[wib][ic];
                s += r * votes[ic * VOTE_STR + c];
            }
            pre[q] = s + bias_l[q];
        }
#pragma unroll
        for (int q = 0; q < 4; ++q) {
            float n2 = pre[q] * pre[q];
            n2 += __shfl_xor(n2, 1, 32);
            n2 += __shfl_xor(n2, 2, 32);
            n2 += __shfl_xor(n2, 4, 32);
            float scale = (n2 / (1.f + n2)) / sqrtf(n2 + CAPS_EPS);
            out[(size_t)pixel * 128 + q * 32 + lane] = pre[q] * scale;
        }
    }
}

extern "C" void kernel_launch(void* const* d_in, const int* in_sizes, int n_in,
                              void* d_out, int out_size, void* d_ws, size_t ws_size,
                              hipStream_t stream) {
    (void)in_sizes; (void)n_in; (void)out_size; (void)ws_size;
    const float* input = (const float*)d_in[0];   // [16,32,32,16,8]
    const float* Wg    = (const float*)d_in[1];   // [3,3,8,128]
    const float* bias  = (const float*)d_in[2];   // [1,1,16,8] -> flat [128]
    float* out = (float*)d_out;                   // [16,32,32,16,8]
    float* xTp = (float*)d_ws;                    // ~9 MB scratch: padded transposed input

    int ngroups = 16 * PH * PH * 16;
    caps_pad_transpose_kernel<<<(ngroups + 255) / 256, 256, 0, stream>>>(input, xTp);
    caps_conv_route_kernel<<<NPIX / (2 * PIX_PER_WAVE), 64, 0, stream>>>(xTp, Wg, bias, out);
}